// eied_ctrlr_48292612276330
// MI455X (gfx1250) — compile-verified
//
#include <hip/hip_runtime.h>

typedef float v2f __attribute__((ext_vector_type(2)));
typedef float v8f __attribute__((ext_vector_type(8)));
typedef const __attribute__((address_space(1))) float* gfloat_p;

// ---- CDNA5 fp32 WMMA: D[16x16] = A[16x4] * B[4x16] + C ----
__device__ __forceinline__ v8f wmma4(v2f a, v2f b, v8f c) {
  return __builtin_amdgcn_wmma_f32_16x16x4_f32(false, a, false, b, (short)0, c,
                                               false, false);
}

// 16-lane half swap (lanes 0-15 <-> 16-31) as one VALU op:
// v_permlanex16_b32 with identity lane selects.
__device__ __forceinline__ float sx16(float v) {
  int i = __builtin_bit_cast(int, v);
  int r = __builtin_amdgcn_permlanex16(i, i, 0x76543210, 0xfedcba98, false, false);
  return __builtin_bit_cast(float, r);
}
__device__ __forceinline__ float selhalf(int lane, float lo, float hi) {
  return (lane < 16) ? lo : hi;
}

// Build a B operand (K rows RB..RB+3, N=16 samples) from a D accumulator whose
// VGPR j holds row j (lanes 0-15) and row j+8 (lanes 16-31).
// B layout mirrors A: vgpr0 = {K+0 | K+2}, vgpr1 = {K+1 | K+3}.
template <int RB>
__device__ __forceinline__ v2f makeB(const v8f& D, int lane) {
  v2f b;
  if constexpr (RB < 8) {
    b.x = selhalf(lane, D[RB + 0], sx16(D[RB + 2]));
    b.y = selhalf(lane, D[RB + 1], sx16(D[RB + 3]));
  } else {
    b.x = selhalf(lane, sx16(D[RB - 8]), D[RB - 6]);
    b.y = selhalf(lane, sx16(D[RB - 7]), D[RB - 5]);
  }
  return b;
}

// ---- block-structured concatenated weight element accessors ----
__device__ __forceinline__ float w1elem(const float* W1, int r, int f) {
  if (r >= 24 || f >= 15) return 0.0f;
  int kb = r >> 2, h = r & 3;
  int blk = (kb == 2) ? 2 : (kb & 1);  // branch input: xl,yl,xr,yl,xl,yl
  int lo = blk * 5;
  if (f < lo || f >= lo + 5) return 0.0f;
  return W1[(kb * 4 + h) * 5 + (f - lo)];
}
__device__ __forceinline__ float w2elem(const float* W2, int r, int k) {
  if (r >= 24 || k >= 24) return 0.0f;
  if ((r >> 2) != (k >> 2)) return 0.0f;
  return W2[((r >> 2) * 4 + (r & 3)) * 4 + (k & 3)];
}
__device__ __forceinline__ float w3elem(const float* W3_2, const float* W3_1,
                                        int r, int k) {
  if (r < 8) {
    int head = r >> 1, o = r & 1;
    int hb = (head == 0) ? 0 : (head == 1) ? 4 : (head == 2) ? 16 : 20;
    if (k >= hb && k < hb + 4) return W3_2[(head * 2 + o) * 4 + (k - hb)];
    return 0.0f;
  } else if (r == 8) {
    return (k >= 8 && k < 12) ? W3_1[k - 8] : 0.0f;
  } else if (r == 9) {
    return (k >= 12 && k < 16) ? W3_1[4 + (k - 12)] : 0.0f;
  }
  return 0.0f;
}
__device__ __forceinline__ float b3elem(const float* b3_2, const float* b3_1,
                                        int r) {
  if (r < 8) return b3_2[r];
  if (r == 8) return b3_1[0];
  if (r == 9) return b3_1[1];
  return 0.0f;
}

// All loop-invariant per-lane state. A[0..3]=L1 tile0, A[4..6]=L1 tile1,
// A[7..10]=L2 tile0, A[11..12]=L2 tile1, A[13..18]=L3.
// bs* = biases pre-scaled by -log2(e) (folded into the sigmoid FMA); bv3 raw.
struct LaneConsts {
  v2f A[19];
  float bs1_0[8], bs1_1[8], bs2_0[8], bs2_1[8], bv3[8];
  unsigned long long pf[8];  // per-lane feature base addresses (global AS)
};
#define LANE_REC_BYTES 384  // >= sizeof(LaneConsts)=376, 16B-aligned stride

__device__ __forceinline__ void compute_lane_consts(
    int lane, const float* xl, const float* yl, const float* xr,
    const float* W1, const float* b1, const float* W2, const float* b2,
    const float* W3_2, const float* b3_2, const float* W3_1, const float* b3_1,
    LaneConsts& L) {
  const int m16 = lane & 15;
  const int hi = lane >> 4;
  const float nl2e = -1.44269504088896340f;
#pragma unroll
  for (int t = 0; t < 4; ++t) {
    int kb = 4 * t + hi * 2;
    L.A[t].x = w1elem(W1, m16, kb);
    L.A[t].y = w1elem(W1, m16, kb + 1);
  }
#pragma unroll
  for (int t = 0; t < 3; ++t) {
    int kb = 4 * t + hi * 2;
    L.A[4 + t].x = w1elem(W1, 16 + m16, kb);
    L.A[4 + t].y = w1elem(W1, 16 + m16, kb + 1);
  }
#pragma unroll
  for (int t = 0; t < 4; ++t) {
    int kb = 4 * t + hi * 2;
    L.A[7 + t].x = w2elem(W2, m16, kb);
    L.A[7 + t].y = w2elem(W2, m16, kb + 1);
  }
#pragma unroll
  for (int t = 0; t < 2; ++t) {
    int kb = 16 + 4 * t + hi * 2;
    L.A[11 + t].x = w2elem(W2, 16 + m16, kb);
    L.A[11 + t].y = w2elem(W2, 16 + m16, kb + 1);
  }
#pragma unroll
  for (int t = 0; t < 6; ++t) {
    int kb = 4 * t + hi * 2;
    L.A[13 + t].x = w3elem(W3_2, W3_1, m16, kb);
    L.A[13 + t].y = w3elem(W3_2, W3_1, m16, kb + 1);
  }
#pragma unroll
  for (int j = 0; j < 8; ++j) {
    L.bs1_0[j] = nl2e * (hi ? b1[j + 8] : b1[j]);
    L.bs1_1[j] = nl2e * (hi ? 0.0f : b1[16 + j]);
    L.bs2_0[j] = nl2e * (hi ? b2[j + 8] : b2[j]);
    L.bs2_1[j] = nl2e * (hi ? 0.0f : b2[16 + j]);
    L.bv3[j] = hi ? b3elem(b3_2, b3_1, j + 8) : b3elem(b3_2, b3_1, j);
  }
  // Feature of this lane for (t,j): f = 4t + 2*hi + j; f==15 is zero-padded
  // (duplicate feat 13; masked to 0 in the loop).
#pragma unroll
  for (int t = 0; t < 4; ++t) {
#pragma unroll
    for (int j = 0; j < 2; ++j) {
      int f = 4 * t + hi * 2 + j;
      if (f == 15) f = 13;
      const float* base = (f < 5) ? (xl + f) : (f < 10) ? (yl + (f - 5))
                                             : (xr + (f - 10));
      L.pf[t * 2 + j] = (unsigned long long)(uintptr_t)base;
    }
  }
}

__global__ void __launch_bounds__(32)
setup_blob_kernel(const float* xl, const float* yl, const float* xr,
                  const float* W1, const float* b1, const float* W2,
                  const float* b2, const float* W3_2, const float* b3_2,
                  const float* W3_1, const float* b3_1, char* blob) {
  int lane = threadIdx.x;
  if (lane >= 32) return;
  LaneConsts L;
  compute_lane_consts(lane, xl, yl, xr, W1, b1, W2, b2, W3_2, b3_2, W3_1, b3_1,
                      L);
  *(LaneConsts*)(blob + (size_t)lane * LANE_REC_BYTES) = L;
}

__device__ __forceinline__ v8f act(const v8f& c, const float bs[8]) {
  v8f h;
#pragma unroll
  for (int j = 0; j < 8; ++j) {
    // sigmoid(c+b) = rcp(1 + exp2(-(c+b)*log2 e)); bias pre-scaled into bs.
    float e = __builtin_amdgcn_exp2f(fmaf(c[j], -1.44269504088896340f, bs[j]));
    h[j] = __builtin_amdgcn_rcpf(1.0f + e);
  }
  return h;
}

__global__ void __launch_bounds__(256)
mlp6_wmma_kernel(const float* __restrict__ xl, const float* __restrict__ yl,
                 const float* __restrict__ xr, const float* __restrict__ W1,
                 const float* __restrict__ b1, const float* __restrict__ W2,
                 const float* __restrict__ b2, const float* __restrict__ W3_2,
                 const float* __restrict__ b3_2, const float* __restrict__ W3_1,
                 const float* __restrict__ b3_1, const float* __restrict__ W4,
                 const float* __restrict__ b4, const char* __restrict__ blob,
                 float* __restrict__ out, int n) {
  const int lane = threadIdx.x & 31;
  const int wave = (int)((blockIdx.x * blockDim.x + threadIdx.x) >> 5);
  const int nwaves = (int)((gridDim.x * blockDim.x) >> 5);
  const int m16 = lane & 15;
  const int hi = lane >> 4;
  const int ntiles = (n + 15) >> 4;

  LaneConsts L;
  if (blob) {  // uniform branch: fast preamble = ~25 wide loads
    L = *(const LaneConsts*)(blob + (size_t)lane * LANE_REC_BYTES);
  } else {
    compute_lane_consts(lane, xl, yl, xr, W1, b1, W2, b2, W3_2, b3_2, W3_1,
                        b3_1, L);
  }
  const float w40_0 = W4[0], w40_1 = W4[1], w40_2 = W4[2], w40_3 = W4[3];
  const float w41_0 = W4[4], w41_1 = W4[5], w41_2 = W4[6], w41_3 = W4[7];
  const float b40 = b4[0], b41 = b4[1];

  for (int tile = wave; tile < ntiles; tile += nwaves) {
    const int s = (tile << 4) + m16;
    const int sc = (s < n) ? s : (n - 1);  // clamp keeps EXEC full for WMMA
    const unsigned long long boff = (unsigned long long)((unsigned)sc * 20u);

    // ---- layer 1 B operands: branchless global-AS per-lane gathers ----
    float fv[8];
#pragma unroll
    for (int k = 0; k < 8; ++k)
      fv[k] = *(gfloat_p)(uintptr_t)(L.pf[k] + boff);  // global_load_b32
    v2f Bx[4];
    Bx[0].x = fv[0]; Bx[0].y = fv[1];
    Bx[1].x = fv[2]; Bx[1].y = fv[3];
    Bx[2].x = fv[4]; Bx[2].y = fv[5];
    Bx[3].x = fv[6]; Bx[3].y = hi ? 0.0f : fv[7];  // K=15 is zero pad

    v8f c10 = {}, c11 = {};
#pragma unroll
    for (int t = 0; t < 4; ++t) c10 = wmma4(L.A[t], Bx[t], c10);
#pragma unroll
    for (int t = 0; t < 3; ++t) c11 = wmma4(L.A[4 + t], Bx[t], c11);
    v8f H10 = act(c10, L.bs1_0);  // h1 rows 0-15
    v8f H11 = act(c11, L.bs1_1);  // h1 rows 16-23 (+pad)

    // ---- layer 2 (block diagonal: only matching K blocks) ----
    v8f c20 = {}, c21 = {};
    c20 = wmma4(L.A[7], makeB<0>(H10, lane), c20);
    c20 = wmma4(L.A[8], makeB<4>(H10, lane), c20);
    c20 = wmma4(L.A[9], makeB<8>(H10, lane), c20);
    c20 = wmma4(L.A[10], makeB<12>(H10, lane), c20);
    c21 = wmma4(L.A[11], makeB<0>(H11, lane), c21);
    c21 = wmma4(L.A[12], makeB<4>(H11, lane), c21);
    v8f H20 = act(c20, L.bs2_0);  // h2 rows 0-15
    v8f H21 = act(c21, L.bs2_1);  // h2 rows 16-23 (+pad)

    // ---- layer 3: all fc3 heads as one 10x24 matrix ----
    v8f c3 = {};
    c3 = wmma4(L.A[13], makeB<0>(H20, lane), c3);
    c3 = wmma4(L.A[14], makeB<4>(H20, lane), c3);
    c3 = wmma4(L.A[15], makeB<8>(H20, lane), c3);
    c3 = wmma4(L.A[16], makeB<12>(H20, lane), c3);
    c3 = wmma4(L.A[17], makeB<0>(H21, lane), c3);
    c3 = wmma4(L.A[18], makeB<4>(H21, lane), c3);

    float o[8];
#pragma unroll
    for (int j = 0; j < 8; ++j) o[j] = c3[j] + L.bv3[j];

    // fc4 on VALU (rows 0-7 live in lanes 0-15; hi lanes compute junk, unused)
    float fxl = fmaf(w40_0, o[0], fmaf(w40_1, o[1],
                fmaf(w40_2, o[4], fmaf(w40_3, o[5], b40))));
    float fyl = fmaf(w41_0, o[2], fmaf(w41_1, o[3],
                fmaf(w41_2, o[6], fmaf(w41_3, o[7], b41))));

    // Full-wave b64 store: hi lanes hold o1 heads (rows 8,9) -> out[+0,+1];
    // lo lanes hold fc4 results -> out[+2,+3]. 16B per sample total.
    v2f r2;
    r2.x = hi ? o[0] : fxl;
    r2.y = hi ? o[1] : fyl;
    float* dst = out + (size_t)s * 4 + (hi ? 0 : 2);
    if (s < n) *reinterpret_cast<v2f*>(dst) = r2;
  }
}

extern "C" void kernel_launch(void* const* d_in, const int* in_sizes, int n_in,
                              void* d_out, int out_size, void* d_ws,
                              size_t ws_size, hipStream_t stream) {
  (void)n_in; (void)out_size;
  const float* xl = (const float*)d_in[0];
  const float* yl = (const float*)d_in[1];
  const float* xr = (const float*)d_in[2];
  // d_in[3] = yr: dead in the reference forward
  const float* W1 = (const float*)d_in[4];
  const float* b1 = (const float*)d_in[5];
  const float* W2 = (const float*)d_in[6];
  const float* b2 = (const float*)d_in[7];
  const float* W3_2 = (const float*)d_in[8];
  const float* b3_2 = (const float*)d_in[9];
  const float* W3_1 = (const float*)d_in[10];
  const float* b3_1 = (const float*)d_in[11];
  const float* W4 = (const float*)d_in[12];
  const float* b4 = (const float*)d_in[13];
  float* out = (float*)d_out;

  const int n = in_sizes[0] / 5;
  if (n <= 0) return;

  const size_t blob_need = (size_t)32 * LANE_REC_BYTES;
  char* blob = nullptr;
  if (d_ws && ws_size >= blob_need) {
    blob = (char*)d_ws;
    setup_blob_kernel<<<1, 32, 0, stream>>>(xl, yl, xr, W1, b1, W2, b2, W3_2,
                                            b3_2, W3_1, b3_1, blob);
  }

  const int ntiles = (n + 15) / 16;
  const int threads = 256;            // 8 waves per block (wave32)
  int blocks = (ntiles + 7) / 8;      // one wave per tile if small
  if (blocks > 8192) blocks = 8192;   // persistent grid-stride otherwise

  mlp6_wmma_kernel<<<blocks, threads, 0, stream>>>(
      xl, yl, xr, W1, b1, W2, b2, W3_2, b3_2, W3_1, b3_1, W4, b4, blob, out, n);
}